// ADDLossSingleManifoldMode_52312701665735
// MI455X (gfx1250) — compile-verified
//
#include <hip/hip_runtime.h>
#include <math.h>

#define BATCH 8192
#define NPTS  1024
#define TILES_PER_BATCH (NPTS / 16)   // 64 tiles of 16 points

typedef __attribute__((ext_vector_type(2))) float v2f;
typedef __attribute__((ext_vector_type(8))) float v8f;

// R = Rx(a) * Ry(b) * Rz(c), row-major R[j*3+k]; result[k] = sum_j p[j]*R[j][k]
__device__ __forceinline__ void euler_R(float a, float b, float c, float R[9]) {
    float ca = cosf(a), sa = sinf(a);
    float cb = cosf(b), sb = sinf(b);
    float cc = cosf(c), sc = sinf(c);
    R[0] = cb * cc;               R[1] = -cb * sc;              R[2] = sb;
    R[3] = ca * sc + sa * sb * cc; R[4] = ca * cc - sa * sb * sc; R[5] = -sa * cb;
    R[6] = sa * sc - ca * sb * cc; R[7] = sa * cc + ca * sb * sc; R[8] = ca * cb;
}

__global__ void zero_out_kernel(float* out) { out[0] = 0.0f; }

__global__ void finish_kernel(float* out) {
    out[0] *= (1.0f / ((float)BATCH * (float)NPTS));   // mean over B*N
}

__global__ __launch_bounds__(256) void add_loss_main(
    const float* __restrict__ pred,   // (B,4)
    const float* __restrict__ mode,   // (B,)
    const float* __restrict__ gt,     // (B,3)
    const float* __restrict__ point,  // (B,N,3)
    float* __restrict__ out)          // scalar accumulator
{
    const int tid  = blockIdx.x * blockDim.x + threadIdx.x;
    const int b    = tid >> 5;          // one wave32 per batch row
    const int lane = threadIdx.x & 31;
    const int m    = lane & 15;
    const bool lo  = lane < 16;

    // ---- manifold2euler (redundant per lane; negligible vs memory time) ----
    const float m1 = pred[b * 4 + 0];
    const float m2 = pred[b * 4 + 1];
    const float m3 = pred[b * 4 + 2];
    const float m4 = pred[b * 4 + 3];
    const float sgn = (mode[b] > 0.5f) ? 1.0f : -1.0f;
    float e2 = sgn * asinf(sqrtf(m3 * m3 / (m1 * m1 + m2 * m2 + m3 * m3)));
    float e3 = atan2f(m4, m3 / (sinf(e2) + 1e-9f));
    float tc = cosf(e2) * cosf(e3);
    float e1 = atan2f(m2 / tc, m1 / tc);
    if (!(e3 > 0.0f)) e3 += 6.283185307179586477f;

    float Rp[9], Rg[9];
    euler_R(e1, e2, e3, Rp);
    euler_R(gt[b * 3 + 0], gt[b * 3 + 1], gt[b * 3 + 2], Rg);

    const float d00 = Rp[0] - Rg[0], d01 = Rp[1] - Rg[1], d02 = Rp[2] - Rg[2];
    const float d10 = Rp[3] - Rg[3], d11 = Rp[4] - Rg[4], d12 = Rp[5] - Rg[5];
    const float d20 = Rp[6] - Rg[6], d21 = Rp[7] - Rg[7], d22 = Rp[8] - Rg[8];

    // ---- A-matrix (16x4 f32): A[M][K] = D[K][M], rows M>=3 zero ----
    // lanes 0-15: V0=A[M][0], V1=A[M][1]; lanes 16-31: V0=A[M][2], V1=A[M][3]=0
    float a0 = 0.0f, a1 = 0.0f;
    if (lo) {
        a0 = (m == 0) ? d00 : (m == 1) ? d01 : (m == 2) ? d02 : 0.0f;
        a1 = (m == 0) ? d10 : (m == 1) ? d11 : (m == 2) ? d12 : 0.0f;
    } else {
        a0 = (m == 0) ? d20 : (m == 1) ? d21 : (m == 2) ? d22 : 0.0f;
    }
    const v2f Av = {a0, a1};

    // ---- stream 64 tiles of 16 points: C = A(=D^T, 16x4) x B(=P^T, 4x16) ----
    const float* __restrict__ pbase = point + (size_t)b * (NPTS * 3);
    const int j0 = lo ? 0 : 2;          // lower lanes load x, upper lanes load z
    float acc = 0.0f;

    for (int t = 0; t < TILES_PER_BATCH; ++t) {
        const float* pp = pbase + (size_t)(t * 16 + m) * 3;
        if (t + 1 < TILES_PER_BATCH)
            __builtin_prefetch(pp + 48, 0, 3);   // next tile, near-cache fill

        // B-matrix (4x16 f32): V0 = row K0 (lo) / K2 (hi); V1 = row K1 (lo) / K3=0 (hi)
        float b0 = pp[j0];
        float b1 = pp[1];
        if (!lo) b1 = 0.0f;
        const v2f Bv = {b0, b1};

        v8f C = {};
        C = __builtin_amdgcn_wmma_f32_16x16x4_f32(
                /*neg_a=*/false, Av, /*neg_b=*/false, Bv,
                /*c_mod=*/(short)0, C, /*reuse_a=*/false, /*reuse_b=*/false);

        // C[r] holds row M=r (lanes 0-15, N=lane) and M=r+8 (lanes 16-31).
        // Rows 0/1/2 = dx/dy/dz of this lane's point; rows 8-10 are zero,
        // so upper-half lanes naturally contribute 0.
        const float dx = C[0], dy = C[1], dz = C[2];
        // bare v_sqrt_f32 (~1 ulp) instead of the IEEE-exact libcall expansion
        acc += __builtin_amdgcn_sqrtf(dx * dx + dy * dy + dz * dz);
    }

    // ---- wave32 reduction, one atomic per wave ----
    for (int off = 16; off >= 1; off >>= 1)
        acc += __shfl_down(acc, off, 32);
    if (lane == 0)
        atomicAdd(out, acc);
}

extern "C" void kernel_launch(void* const* d_in, const int* in_sizes, int n_in,
                              void* d_out, int out_size, void* d_ws, size_t ws_size,
                              hipStream_t stream) {
    const float* pred  = (const float*)d_in[0];
    const float* mode  = (const float*)d_in[1];
    const float* gt    = (const float*)d_in[2];
    const float* point = (const float*)d_in[3];
    float* out = (float*)d_out;

    zero_out_kernel<<<1, 1, 0, stream>>>(out);

    // one wave32 per batch: 8192 waves * 32 lanes = 262144 threads
    dim3 block(256);
    dim3 grid((BATCH * 32) / 256);
    add_loss_main<<<grid, block, 0, stream>>>(pred, mode, gt, point, out);

    finish_kernel<<<1, 1, 0, stream>>>(out);
}